// Z_78228534329909
// MI455X (gfx1250) — compile-verified
//
#include <hip/hip_runtime.h>

// reference: out = diag(ics_mask - params * r_mask), N = 12288, f32 output.
// Pure store-bandwidth problem: 604 MB of output, ~26 us floor at 23.3 TB/s.
// Strategy: one block per row, 256 threads (8 wave32s), each thread does 12
// unrolled 128-bit non-temporal stores (bypass the 192 MB L2 -- output is 3x
// larger than L2, streaming writes must not thrash it).

#define NDIM   12288
#define BLOCK  256
#define QROW   (NDIM / 4)          // 3072 float4 per row
#define ITERS  (QROW / BLOCK)      // 12 float4 stores per thread

typedef __attribute__((ext_vector_type(4))) float f32x4;

__global__ __launch_bounds__(BLOCK) void diag_fill_kernel(
    const float* __restrict__ params,
    const int*   __restrict__ r_mask,
    const int*   __restrict__ ics_mask,
    float*       __restrict__ out)
{
    const int row = blockIdx.x;

    // Block-uniform loads -> scalar loads; diagonal value for this row.
    const float dv = (float)ics_mask[row] - params[row] * (float)r_mask[row];

    const int dq = row >> 2;   // which float4 in this row holds the diagonal
    const int dl = row & 3;    // which lane of that float4

    f32x4* rowp = (f32x4*)(out + (size_t)row * (size_t)NDIM);

#pragma unroll
    for (int k = 0; k < ITERS; ++k) {
        const int q = (int)threadIdx.x + k * BLOCK;
        f32x4 v;
        // Branchless: zeros everywhere, diagonal value in exactly one lane of
        // one float4 per row. A few v_cndmask per store, negligible vs VMEM.
        v.x = (q == dq && dl == 0) ? dv : 0.0f;
        v.y = (q == dq && dl == 1) ? dv : 0.0f;
        v.z = (q == dq && dl == 2) ? dv : 0.0f;
        v.w = (q == dq && dl == 3) ? dv : 0.0f;
        // Non-temporal 128-bit store: global_store_b128 with TH=NT on gfx1250.
        __builtin_nontemporal_store(v, rowp + q);
    }
}

extern "C" void kernel_launch(void* const* d_in, const int* in_sizes, int n_in,
                              void* d_out, int out_size, void* d_ws, size_t ws_size,
                              hipStream_t stream)
{
    const float* params   = (const float*)d_in[0];
    const int*   r_mask   = (const int*)d_in[1];
    const int*   ics_mask = (const int*)d_in[2];
    float*       out      = (float*)d_out;

    // One block per output row; writes the entire 12288x12288 matrix each call.
    diag_fill_kernel<<<NDIM, BLOCK, 0, stream>>>(params, r_mask, ics_mask, out);
}